// MessagePassingLayer_28080496181525
// MI455X (gfx1250) — compile-verified
//
#include <hip/hip_runtime.h>
#include <hip/hip_bf16.h>

// CDNA5 / gfx1250, wave32. f32 WMMA path (V_WMMA_F32_16X16X4_F32).
typedef float v2f __attribute__((ext_vector_type(2)));
typedef float v8f __attribute__((ext_vector_type(8)));

#define K_NB    32
#define D_IN    128
#define F_DIMM  32
#define OUT_DIM 128
#define TILE_N  16
#define LDS_STRIDE 132   // 128 + 4 pad: conflict-free strided row reads, 16B-aligned f4 stores

// Single-use streaming data: non-temporal (TH=NT) so the 819MB h_neighbors
// stream does not evict the reused W_i / diff / W_gamma working set from L2.
#define NT_LOAD(p)      __builtin_nontemporal_load(p)
#define NT_STORE(p, v)  __builtin_nontemporal_store((v), (p))

__global__ __launch_bounds__(256)
void msg_layer_kernel(const float* __restrict__ h_center,    // [N,128]
                      const float* __restrict__ h_neighbors, // [N,32,128]
                      const float* __restrict__ diff,        // [N,32,32]
                      const float* __restrict__ W_i,         // [128,128]
                      const float* __restrict__ b_i,         // [128]
                      const float* __restrict__ W_gamma,     // [32,128]
                      float* __restrict__ out)               // [N,128]
{
    __shared__ float lds_hc[TILE_N * LDS_STRIDE];

    const int tid  = threadIdx.x;
    const int wave = tid >> 5;      // 8 waves: o-tile id
    const int lane = tid & 31;
    const int col  = lane & 15;     // N-index within 16x16 tile
    const int half = lane >> 4;     // upper/lower half-wave
    const int o    = wave * 16 + col;
    const int n0   = blockIdx.x * TILE_N;

    // ---- stage h_center tile [16 x 128] into LDS (coalesced b128 loads) ----
    {
        const float4* src = (const float4*)(h_center + (size_t)n0 * D_IN);
        for (int c = tid; c < TILE_N * (D_IN / 4); c += 256) {
            int row = c >> 5;            // /(128/4)
            int c4  = c & 31;
            float4 v = src[row * 32 + c4];
            float* dst = &lds_hc[row * LDS_STRIDE + c4 * 4];
            dst[0] = v.x; dst[1] = v.y; dst[2] = v.z; dst[3] = v.w;
        }
    }
    __syncthreads();

    // ---- Phase A: hc = h_center @ W_i  (16x128x128 per block, 16x16 per wave) ----
    v8f hc_acc = {0.f, 0.f, 0.f, 0.f, 0.f, 0.f, 0.f, 0.f};
    {
        const float* aRow = &lds_hc[col * LDS_STRIDE];
        #pragma unroll
        for (int k0 = 0; k0 < D_IN; k0 += 4) {
            const int ka = k0 + half * 2;
            v2f a; a.x = aRow[ka];
                   a.y = aRow[ka + 1];
            v2f b; b.x = W_i[ka * OUT_DIM + o];
                   b.y = W_i[(ka + 1) * OUT_DIM + o];
            hc_acc = __builtin_amdgcn_wmma_f32_16x16x4_f32(
                false, a, false, b, (short)0, hc_acc, false, false);
        }
    }

    // Park hc (+bias) in LDS so Phase C can read it by (n,o) without
    // variable-index VGPR extraction.
    const float bias = b_i[o];
    __syncthreads();
    #pragma unroll
    for (int v = 0; v < 8; ++v)
        lds_hc[(v + 8 * half) * LDS_STRIDE + o] = hc_acc[v] + bias;
    __syncthreads();

    // ---- Preload W_gamma column entries this lane needs ----
    // lane covers f in { ft*16 + half*8 + v : v=0..7, ft=0..1 }
    float wg[16];
    #pragma unroll
    for (int ft = 0; ft < 2; ++ft)
        #pragma unroll
        for (int v = 0; v < 8; ++v)
            wg[ft * 8 + v] = W_gamma[(ft * 16 + half * 8 + v) * OUT_DIM + o];

    // ---- Phase B: per node, t = diff^T (32f x 32k) @ hn (32k x 128o) via WMMA;
    //      msg[o] = sum_f Wg[f,o] * t[f,o]  (hn streams once as the B operand) ----
    const size_t hnBase = (size_t)n0 * K_NB * OUT_DIM;
    const size_t dfBase = (size_t)n0 * K_NB * F_DIMM;

    #pragma unroll
    for (int n = 0; n < TILE_N; ++n) {
        const float* hn = h_neighbors + hnBase + (size_t)n * (K_NB * OUT_DIM);
        const float* df = diff        + dfBase + (size_t)n * (K_NB * F_DIMM);

        float msgp = 0.f;
        #pragma unroll
        for (int ft = 0; ft < 2; ++ft) {
            v8f c = {0.f, 0.f, 0.f, 0.f, 0.f, 0.f, 0.f, 0.f};
            #pragma unroll
            for (int k0 = 0; k0 < K_NB; k0 += 4) {
                const int ka = k0 + half * 2;
                // A = diff^T: A[f][k] = df[k*32 + f]; M = ft*16 + col
                // (lanes contiguous in f; reused by all 8 waves -> regular TH)
                v2f a; a.x = df[ka * F_DIMM + ft * 16 + col];
                       a.y = df[(ka + 1) * F_DIMM + ft * 16 + col];
                // B = hn: B[k][o] (lanes contiguous in o -> 64B segments);
                // single-use stream -> non-temporal
                v2f b; b.x = NT_LOAD(&hn[ka * OUT_DIM + o]);
                       b.y = NT_LOAD(&hn[(ka + 1) * OUT_DIM + o]);
                c = __builtin_amdgcn_wmma_f32_16x16x4_f32(
                    false, a, false, b, (short)0, c, false, false);
            }
            #pragma unroll
            for (int v = 0; v < 8; ++v)
                msgp += wg[ft * 8 + v] * c[v];
        }
        // combine the two half-wave f partitions (same column o in lane L and L^16)
        msgp += __shfl_xor(msgp, 16, 32);

        // ---- Phase C: silu(hc + msg) ----
        const float x   = lds_hc[n * LDS_STRIDE + o] + msgp;
        const float res = x / (1.f + expf(-x));
        if (half == 0)
            NT_STORE(&out[(size_t)(n0 + n) * OUT_DIM + o], res);
    }
}

extern "C" void kernel_launch(void* const* d_in, const int* in_sizes, int n_in,
                              void* d_out, int out_size, void* d_ws, size_t ws_size,
                              hipStream_t stream) {
    const float* h_center    = (const float*)d_in[0];
    const float* h_neighbors = (const float*)d_in[1];
    const float* diff        = (const float*)d_in[2];
    const float* W_i         = (const float*)d_in[3];
    const float* b_i         = (const float*)d_in[4];
    const float* W_gamma     = (const float*)d_in[5];
    float* out = (float*)d_out;

    const int n_total = in_sizes[0] / D_IN;       // 50000 (divisible by TILE_N=16)
    const int grid    = (n_total + TILE_N - 1) / TILE_N;

    msg_layer_kernel<<<grid, 256, 0, stream>>>(h_center, h_neighbors, diff,
                                               W_i, b_i, W_gamma, out);
    (void)n_in; (void)out_size; (void)d_ws; (void)ws_size;
}